// VDEmbedding_29102698397779
// MI455X (gfx1250) — compile-verified
//
#include <hip/hip_runtime.h>

// Variational-dropout embedding: out[t, :] = (x[t] != 0) ? raw_weight[x[t], :] * mask[x[t]] : 0
// Pure gather+scale, memory-bound. One wave32 per token row (512 f32 = 2KB):
// each lane moves 4 x float4 via global_load_b128 / nontemporal global_store_b128.
// Index + mask loads are wave-uniform -> scalar (s_load). Next row is prefetched
// with global_prefetch_b8 while the current row streams through.

typedef float v4f __attribute__((ext_vector_type(4)));

#define EMB_DIM 512
#define PAD_IDX 0
#define V4_PER_ROW (EMB_DIM / 4)        /* 128 */
#define V4_PER_LANE (V4_PER_ROW / 32)   /* 4   */

__global__ void vd_embed_kernel(const int* __restrict__ x,
                                const float* __restrict__ w,
                                const float* __restrict__ mask,
                                float* __restrict__ out,
                                int n_tokens) {
    const int lane = (int)(threadIdx.x & 31u);
    int wave = (int)((blockIdx.x * blockDim.x + threadIdx.x) >> 5);
    const int n_waves = (int)((gridDim.x * blockDim.x) >> 5);
    // Token id is uniform across the wave: make it an SGPR so the index and
    // mask loads become scalar loads.
    wave = __builtin_amdgcn_readfirstlane(wave);

    for (int t = wave; t < n_tokens; t += n_waves) {
        const int idx = x[t];                       // uniform -> s_load_b32

        // Prefetch the next row this wave will process: 32 lanes x 64B = 2KB row.
        const int tn = t + n_waves;
        if (tn < n_tokens) {
            const int idxn = x[tn];                 // uniform -> s_load_b32
            const char* p = (const char*)(w + (size_t)idxn * EMB_DIM);
            __builtin_prefetch(p + (size_t)lane * 64, /*rw=*/0, /*loc=*/3);
        }

        float m = mask[idx];                        // uniform -> s_load_b32
        m = (idx == PAD_IDX) ? 0.0f : m;            // reference's where() guard

        const v4f* __restrict__ src = (const v4f*)(w + (size_t)idx * EMB_DIM);
        v4f* __restrict__ dst = (v4f*)(out + (size_t)t * EMB_DIM);

        v4f r[V4_PER_LANE];
#pragma unroll
        for (int i = 0; i < V4_PER_LANE; ++i)
            r[i] = src[lane + 32 * i];              // 4x global_load_b128, coalesced

#pragma unroll
        for (int i = 0; i < V4_PER_LANE; ++i) {
            v4f v = r[i] * m;
            // Output is write-once streaming: non-temporal keeps the 103MB
            // weight table resident in the 192MB L2.
            __builtin_nontemporal_store(v, &dst[lane + 32 * i]);
        }
    }
}

extern "C" void kernel_launch(void* const* d_in, const int* in_sizes, int n_in,
                              void* d_out, int out_size, void* d_ws, size_t ws_size,
                              hipStream_t stream) {
    const int*   x    = (const int*)d_in[0];    // [8, 2048] token ids
    const float* w    = (const float*)d_in[1];  // [50257, 512] raw weights
    const float* mask = (const float*)d_in[2];  // [50257] dropout row mask
    float*       out  = (float*)d_out;          // [8, 2048, 512]

    const int n_tokens = in_sizes[0];           // 16384

    const int block = 256;                      // 8 wave32 per block
    const int tokens_per_wave = 4;
    const int waves_per_block = block / 32;
    int blocks = (n_tokens + waves_per_block * tokens_per_wave - 1) /
                 (waves_per_block * tokens_per_wave);
    if (blocks < 1) blocks = 1;

    vd_embed_kernel<<<blocks, block, 0, stream>>>(x, w, mask, out, n_tokens);
}